// MultiHeadTransposedAttention_30777735643434
// MI455X (gfx1250) — compile-verified
//
#include <hip/hip_runtime.h>
#include <stdint.h>

// ---------------------------------------------------------------------------
// MultiHeadTransposedAttention (XCA) for MI455X / gfx1250, wave32 + WMMA f32.
// qkv: [4, 576, 256, 256] f32 ; temperature: [6] f32 ; out: [4, 192, 256, 256]
// Memory-bound (~0.8 GB moved, ~6.4 GFLOP): read qkv once, write out once.
// Phase 1 uses GLOBAL_LOAD_ASYNC_TO_LDS_B128 (ASYNCcnt) + double buffering.
// ---------------------------------------------------------------------------

typedef __attribute__((ext_vector_type(2))) float v2f;
typedef __attribute__((ext_vector_type(8))) float v8f;
typedef int v4i_gcc __attribute__((vector_size(16)));   // matches builtin param

#define HW      65536      // 256*256 spatial
#define NHEADS  6
#define CH      32         // channels per head
#define C3      576
#define CC      192        // total channels per q/k/v
#define BH      24         // b * heads
#define NT      64         // LDS tile columns (phase 1)
#define LSTR    68         // padded stride: 68 % 64 == 4 -> conflict-free b64 frag reads
#define TILE_F  (CH * LSTR)              // 2176 floats per q or k tile
#define NCHUNK1 32         // n-chunks per (b,h) in phase 1
#define CHUNK1  (HW / NCHUNK1)           // 2048 cols per workgroup
#define TILES1  (CHUNK1 / NT)            // 32 double-buffered tiles per WG
#define NCHUNK3 32
#define CHUNK3  (HW / NCHUNK3)           // 2048 cols per workgroup (phase 3)

// D = A(16x4) * B(4x16) + C, all f32, wave32.
__device__ __forceinline__ v8f wmma4(v2f a, v2f b, v8f c) {
  return __builtin_amdgcn_wmma_f32_16x16x4_f32(false, a, false, b, (short)0, c,
                                               false, false);
}

// Async global -> LDS 16B copy (ASYNCcnt).  Builtin wants int4* in AS(1)/AS(3);
// the LDS operand is the 32-bit LDS byte offset (low 32 bits of generic ptr).
__device__ __forceinline__ void async_copy_b128(const float* g, float* l) {
#if defined(__has_builtin) && __has_builtin(__builtin_amdgcn_global_load_async_to_lds_b128)
  typedef __attribute__((address_space(1))) v4i_gcc* gp_t;
  typedef __attribute__((address_space(3))) v4i_gcc* lp_t;
  __builtin_amdgcn_global_load_async_to_lds_b128(
      (gp_t)(uintptr_t)g, (lp_t)(uint32_t)(uintptr_t)l, 0, 0);
#else
  asm volatile("global_load_async_to_lds_b128 %0, %1, off"
               :: "v"((uint32_t)(uintptr_t)l), "v"(g)
               : "memory");
#endif
}

__device__ __forceinline__ void wait_async0() {
#if defined(__has_builtin) && __has_builtin(__builtin_amdgcn_s_wait_asynccnt)
  __builtin_amdgcn_s_wait_asynccnt(0);
#else
  asm volatile("s_wait_asynccnt 0x0" ::: "memory");
#endif
}

// ---------------------------------------------------------------------------
// Phase 1: per (b,h,chunk): partial S = q * k^T (32x32) via WMMA,
//          plus partial sum-of-squares per q/k row (from the fragments).
// ---------------------------------------------------------------------------
__global__ __launch_bounds__(256)
void xca_phase1(const float* __restrict__ qkv,
                float* __restrict__ pS,     // [BH*NCHUNK1][1024]
                float* __restrict__ pSsq) { // [BH*NCHUNK1][64] (32 q + 32 k)
  const int tid  = threadIdx.x;
  const int wave = tid >> 5;
  const int lane = tid & 31;
  const int lr   = lane & 15;   // M/N index within a 16-tile
  const int hi   = lane >> 4;   // K half-select

  const int bh    = blockIdx.x / NCHUNK1;
  const int chunk = blockIdx.x % NCHUNK1;
  const int b = bh / NHEADS, h = bh % NHEADS;

  const float* qbase = qkv + ((size_t)b * C3 + (size_t)h * CH) * HW;
  const float* kbase = qbase + (size_t)CC * HW;

  // 2 buffers x (q tile + k tile); reused later as 8x1024 reduction scratch.
  __shared__ alignas(16) float smem[4 * TILE_F];   // 8704 floats (>= 8192)
  __shared__ float sSsq[64];
  if (tid < 64) sSsq[tid] = 0.0f;

  // async loader mapping: each thread owns one row, 2x b128 per tile per matrix
  const int lrow = tid >> 3;    // 0..31
  const int lj   = tid & 7;     // 0..7
  const int colBase = chunk * CHUNK1;

  v8f acc[2][2] = {{{}, {}}, {{}, {}}};
  float sq_lo = 0.f, sq_hi = 0.f, sk_lo = 0.f, sk_hi = 0.f;

  // prologue: issue tile 0 into buffer 0
  {
    float* dQ = smem + 0 * TILE_F;
    float* dK = smem + 1 * TILE_F;
    const size_t rowg = (size_t)lrow * HW + colBase;
#pragma unroll
    for (int i = 0; i < 2; ++i) {
      const int c = i * 32 + lj * 4;
      async_copy_b128(qbase + rowg + c, dQ + lrow * LSTR + c);
      async_copy_b128(kbase + rowg + c, dK + lrow * LSTR + c);
    }
  }

  for (int t = 0; t < TILES1; ++t) {
    const int buf = t & 1;
    wait_async0();        // this wave's tile-t copies landed in LDS
    __syncthreads();      // everyone's landed; everyone done reading tile t-1

    if (t + 1 < TILES1) { // issue tile t+1 into the other buffer
      float* dQ = smem + ((t + 1) & 1) * 2 * TILE_F;
      float* dK = dQ + TILE_F;
      const size_t rowg = (size_t)lrow * HW + colBase + (t + 1) * NT;
#pragma unroll
      for (int i = 0; i < 2; ++i) {
        const int c = i * 32 + lj * 4;
        async_copy_b128(qbase + rowg + c, dQ + lrow * LSTR + c);
        async_copy_b128(kbase + rowg + c, dK + lrow * LSTR + c);
      }
    }

    const float* sQ = smem + buf * 2 * TILE_F;
    const float* sK = sQ + TILE_F;
    // each wave owns an 8-column slice -> 2 K-chunks of 4
#pragma unroll
    for (int kk = 0; kk < 2; ++kk) {
      const int c0 = wave * 8 + kk * 4 + hi * 2;
      // A(q) and B(k) fragments share the same lane->(row, k) map (ISA 7.12.2)
      v2f aq0 = *(const v2f*)&sQ[(0  + lr) * LSTR + c0];
      v2f aq1 = *(const v2f*)&sQ[(16 + lr) * LSTR + c0];
      v2f bk0 = *(const v2f*)&sK[(0  + lr) * LSTR + c0];
      v2f bk1 = *(const v2f*)&sK[(16 + lr) * LSTR + c0];
      // every tile element appears in exactly one fragment across the WG:
      // accumulate sum-of-squares for the L2 norms right here.
      sq_lo += aq0.x * aq0.x + aq0.y * aq0.y;   // q row lr
      sq_hi += aq1.x * aq1.x + aq1.y * aq1.y;   // q row 16+lr
      sk_lo += bk0.x * bk0.x + bk0.y * bk0.y;   // k row lr
      sk_hi += bk1.x * bk1.x + bk1.y * bk1.y;   // k row 16+lr
      acc[0][0] = wmma4(aq0, bk0, acc[0][0]);
      acc[0][1] = wmma4(aq0, bk1, acc[0][1]);
      acc[1][0] = wmma4(aq1, bk0, acc[1][0]);
      acc[1][1] = wmma4(aq1, bk1, acc[1][1]);
    }
  }

  atomicAdd(&sSsq[lr],      sq_lo);   // ds_add_f32
  atomicAdd(&sSsq[16 + lr], sq_hi);
  atomicAdd(&sSsq[32 + lr], sk_lo);
  atomicAdd(&sSsq[48 + lr], sk_hi);

  __syncthreads();                     // frag reads + atomics done
  // stage each wave's 32x32 partial into LDS: red[wave][m*32+n]
  float* red = smem;
#pragma unroll
  for (int mt = 0; mt < 2; ++mt)
#pragma unroll
    for (int nt = 0; nt < 2; ++nt)
#pragma unroll
      for (int r = 0; r < 8; ++r) {
        const int m = mt * 16 + hi * 8 + r;   // C/D layout: M = r + 8*lane[4]
        const int n = nt * 16 + lr;
        red[wave * 1024 + m * 32 + n] = acc[mt][nt][r];
      }
  __syncthreads();
  // cross-wave reduce and emit partials (coalesced)
#pragma unroll
  for (int i = 0; i < 4; ++i) {
    const int e = tid + i * 256;
    float s = 0.f;
#pragma unroll
    for (int w = 0; w < 8; ++w) s += red[w * 1024 + e];
    pS[((size_t)bh * NCHUNK1 + chunk) * 1024 + e] = s;
  }
  if (tid < 64) pSsq[((size_t)bh * NCHUNK1 + chunk) * 64 + tid] = sSsq[tid];
}

// ---------------------------------------------------------------------------
// Phase 2: reduce partials, apply 1/||.||, temperature, row softmax -> attn.
// ---------------------------------------------------------------------------
__global__ __launch_bounds__(128)
void xca_phase2(const float* __restrict__ pS, const float* __restrict__ pSsq,
                const float* __restrict__ temperature,
                float* __restrict__ attnW) {  // [BH][1024]
  const int bh = blockIdx.x;
  const int h  = bh % NHEADS;
  const int tid = threadIdx.x;
  __shared__ float sAttn[CH * CH];
  __shared__ float sInv[64];

  if (tid < 64) {
    float s = 0.f;
    for (int c = 0; c < NCHUNK1; ++c)
      s += pSsq[((size_t)bh * NCHUNK1 + c) * 64 + tid];
    sInv[tid] = 1.0f / fmaxf(sqrtf(s), 1e-12f);   // matches F.normalize eps
  }
  __syncthreads();

  const float tscale = temperature[h];
#pragma unroll
  for (int i = 0; i < 8; ++i) {
    const int e = tid + i * 128;
    float s = 0.f;
    for (int c = 0; c < NCHUNK1; ++c)
      s += pS[((size_t)bh * NCHUNK1 + c) * 1024 + e];
    const int row = e >> 5, col = e & 31;
    sAttn[e] = s * sInv[row] * sInv[32 + col] * tscale;
  }
  __syncthreads();

  if (tid < 32) {   // one lane per row, softmax over d (32 wide)
    float mx = -INFINITY;
    for (int j = 0; j < 32; ++j) mx = fmaxf(mx, sAttn[tid * 32 + j]);
    float sum = 0.f;
    for (int j = 0; j < 32; ++j) {
      const float ev = __expf(sAttn[tid * 32 + j] - mx);
      sAttn[tid * 32 + j] = ev;
      sum += ev;
    }
    const float inv = 1.0f / sum;
    for (int j = 0; j < 32; ++j) sAttn[tid * 32 + j] *= inv;
  }
  __syncthreads();
#pragma unroll
  for (int i = 0; i < 8; ++i) {
    const int e = tid + i * 128;
    attnW[(size_t)bh * 1024 + e] = sAttn[e];
  }
}

// ---------------------------------------------------------------------------
// Phase 3: out[32, n] = attn[32,32] * v[32, n] via WMMA; pure HBM stream.
// ---------------------------------------------------------------------------
__global__ __launch_bounds__(256)
void xca_phase3(const float* __restrict__ qkv, const float* __restrict__ attnW,
                float* __restrict__ out) {
  const int tid  = threadIdx.x;
  const int wave = tid >> 5;
  const int lane = tid & 31;
  const int lr   = lane & 15;
  const int hi   = lane >> 4;

  const int bh    = blockIdx.x / NCHUNK3;
  const int chunk = blockIdx.x % NCHUNK3;
  const int b = bh / NHEADS, h = bh % NHEADS;

  const float* vbase = qkv + ((size_t)b * C3 + 2 * CC + (size_t)h * CH) * HW;
  float* obase = out + ((size_t)b * CC + (size_t)h * CH) * HW;

  __shared__ alignas(16) float sA[CH * 34];   // stride 34 keeps b64 aligned
  for (int i = tid; i < 1024; i += 256)
    sA[(i >> 5) * 34 + (i & 31)] = attnW[(size_t)bh * 1024 + i];
  __syncthreads();

  // attn fragments are loop-invariant: hold all 16 in registers
  v2f afr[2][8];
#pragma unroll
  for (int mt = 0; mt < 2; ++mt)
#pragma unroll
    for (int kc = 0; kc < 8; ++kc)
      afr[mt][kc] = *(const v2f*)&sA[(mt * 16 + lr) * 34 + kc * 4 + hi * 2];

  const int base = chunk * CHUNK3;
  for (int blk = wave; blk < CHUNK3 / 16; blk += 8) {
    const int n0 = base + blk * 16;
    v8f acc0 = {}, acc1 = {};
#pragma unroll
    for (int kc = 0; kc < 8; ++kc) {
      const float* pv = vbase + (size_t)(kc * 4 + hi * 2) * HW + n0 + lr;
      __builtin_prefetch(pv + 8 * 16, 0, 1);   // next n-block of this wave
      v2f bf;
      bf.x = pv[0];      // B[k][n]   : row kc*4 + hi*2
      bf.y = pv[HW];     // B[k+1][n] : next v row
      acc0 = wmma4(afr[0][kc], bf, acc0);
      acc1 = wmma4(afr[1][kc], bf, acc1);
    }
#pragma unroll
    for (int r = 0; r < 8; ++r) {
      obase[(size_t)(0  + hi * 8 + r) * HW + n0 + lr] = acc0[r];
      obase[(size_t)(16 + hi * 8 + r) * HW + n0 + lr] = acc1[r];
    }
  }
}

// ---------------------------------------------------------------------------
extern "C" void kernel_launch(void* const* d_in, const int* in_sizes, int n_in,
                              void* d_out, int out_size, void* d_ws,
                              size_t ws_size, hipStream_t stream) {
  const float* qkv  = (const float*)d_in[0];
  const float* temp = (const float*)d_in[1];
  float* out = (float*)d_out;

  float* ws    = (float*)d_ws;
  float* pS    = ws;                                   // 24*32*1024 floats
  float* pSsq  = pS + (size_t)BH * NCHUNK1 * 1024;     // 24*32*64 floats
  float* attnW = pSsq + (size_t)BH * NCHUNK1 * 64;     // 24*1024 floats
  // total ws use: ~3.3 MB

  xca_phase1<<<BH * NCHUNK1, 256, 0, stream>>>(qkv, pS, pSsq);
  xca_phase2<<<BH, 128, 0, stream>>>(pS, pSsq, temp, attnW);
  xca_phase3<<<BH * NCHUNK3, 256, 0, stream>>>(qkv, attnW, out);
}